// MultiClsLoss_1082331759381
// MI455X (gfx1250) — compile-verified
//
#include <hip/hip_runtime.h>
#include <stdint.h>

// Problem shape (matches reference setup_inputs)
#define LVL 3
#define BB  32
#define AAA 32768
#define CCC 81
#define NLB (LVL * BB)      // 96
#define NPPC 3
#define MIN_NEGC 100

static constexpr long long TOTAL_ANCH = (long long)LVL * BB * AAA;   // 3,145,728
static constexpr size_t NEGKEY_BYTES = (size_t)TOTAL_ANCH * 4;       // ~12.6 MB

typedef unsigned int       u32;
typedef unsigned long long u64;
typedef u32   uint32x4 __attribute__((ext_vector_type(4)));
typedef int   int32x4  __attribute__((ext_vector_type(4)));
typedef int   int32x8  __attribute__((ext_vector_type(8)));
typedef float v2f      __attribute__((ext_vector_type(2)));
typedef float v8f      __attribute__((ext_vector_type(8)));

// ---------------------------------------------------------------- init
__global__ __launch_bounds__(256) void k_init(float* pos_sum, int* num_pos,
                                              float* totals) {
    int t = threadIdx.x;
    if (t < NLB) { pos_sum[t] = 0.f; num_pos[t] = 0; }
    if (t < 128) totals[t] = 0.f;      // padded to 2 x (16x4) WMMA A-operands
}

// ---------------------------------------------------------------- phase 1
// One wave32 per anchor: lanes cover classes {lane, lane+32, lane+64}.
// Fully coalesced b32 loads (3 x 128B segments per 324B row group).
__global__ __launch_bounds__(256) void k_loss(const float* __restrict__ logits,
                                              const int* __restrict__ labels,
                                              const float* __restrict__ tags,
                                              u32* __restrict__ negkey,
                                              float* __restrict__ pos_sum,
                                              int* __restrict__ num_pos) {
    long long wid = (long long)blockIdx.x * (blockDim.x >> 5) + (threadIdx.x >> 5);
    if (wid >= TOTAL_ANCH) return;
    int lane = threadIdx.x & 31;
    const float* row = logits + wid * CCC;

    float x0 = row[lane];
    float x1 = row[lane + 32];
    bool  has2 = lane < (CCC - 64);                 // lanes 0..16 cover 64..80
    float x2 = has2 ? row[lane + 64] : -__builtin_inff();

    float m = fmaxf(fmaxf(x0, x1), x2);
    #pragma unroll
    for (int off = 16; off; off >>= 1) m = fmaxf(m, __shfl_xor(m, off, 32));

    float s = __expf(x0 - m) + __expf(x1 - m) + (has2 ? __expf(x2 - m) : 0.f);
    #pragma unroll
    for (int off = 16; off; off >>= 1) s += __shfl_xor(s, off, 32);

    int lbl = labels[wid];                          // wave-uniform
    float xl;
    if (lbl < 32)      xl = __shfl(x0, lbl, 32);
    else if (lbl < 64) xl = __shfl(x1, lbl - 32, 32);
    else               xl = __shfl(x2, lbl - 64, 32);

    float loss = m + __logf(s) - xl;                // -log_softmax[label]

    if (lane == 0) {
        float tg = tags[wid];
        u32 key = 0u;
        if (tg == 1.0f) {
            int lb = (int)(wid >> 15);              // A = 32768
            atomicAdd(&pos_sum[lb], loss);
            atomicAdd(&num_pos[lb], 1);
        } else if (tg == -1.0f) {
            key = __float_as_uint(loss);            // loss >= 0 -> bits monotonic
        }
        negkey[wid] = key;                          // 0 == "not a negative"
    }
}

// ---------------------------------------------------------------- reductions
__device__ __forceinline__ float blk_red_f(float v, float* scr) {
    #pragma unroll
    for (int off = 16; off; off >>= 1) v += __shfl_xor(v, off, 32);
    int w = threadIdx.x >> 5, l = threadIdx.x & 31;
    if (l == 0) scr[w] = v;
    __syncthreads();
    if (w == 0) {
        float t = scr[l];                           // blockDim == 1024 -> 32 warps
        #pragma unroll
        for (int off = 16; off; off >>= 1) t += __shfl_xor(t, off, 32);
        if (l == 0) scr[0] = t;
    }
    __syncthreads();
    float r = scr[0];
    __syncthreads();
    return r;
}

__device__ __forceinline__ u32 blk_red_u(u32 v, u32* scr) {
    #pragma unroll
    for (int off = 16; off; off >>= 1) v += __shfl_xor(v, off, 32);
    int w = threadIdx.x >> 5, l = threadIdx.x & 31;
    if (l == 0) scr[w] = v;
    __syncthreads();
    if (w == 0) {
        u32 t = scr[l];
        #pragma unroll
        for (int off = 16; off; off >>= 1) t += __shfl_xor(t, off, 32);
        if (l == 0) scr[0] = t;
    }
    __syncthreads();
    u32 r = scr[0];
    __syncthreads();
    return r;
}

// ---------------------------------------------------------------- phase 2
// One workgroup per (L,B). TDM stages all 32768 keys (128 KB) into LDS once;
// then candidate count/sum + 4-pass MSB radix select + boundary-exact top-k sum,
// all from LDS (CDNA5 320KB LDS makes this possible).
__global__ __launch_bounds__(1024) void k_mine(const u32* __restrict__ negkey,
                                               const float* __restrict__ pos_sum,
                                               const int* __restrict__ num_pos,
                                               float* __restrict__ totals) {
    __shared__ u32   s_keys[AAA];        // 128 KB
    __shared__ u32   s_hist[256];
    __shared__ float s_fred[32];
    __shared__ u32   s_ured[32];
    __shared__ u32   s_bcast[2];

    const int lb = blockIdx.x;
    const u32* gkeys = negkey + (size_t)lb * AAA;

    // --- Tensor Data Mover: DMA 128 KB global -> LDS (one wave issues it) ---
    if (threadIdx.x < 32) {
        u64 gaddr   = (u64)(uintptr_t)gkeys;
        u32 lds_off = (u32)(uintptr_t)(&s_keys[0]);   // LDS aperture: low 32 bits = LDS offset

        // D# group 0: count=1 | lds_addr | global_addr[56:0] | type=2
        uint32x4 g0;
        g0.x = 1u;
        g0.y = lds_off;
        g0.z = (u32)(gaddr & 0xFFFFFFFFull);
        g0.w = (u32)((gaddr >> 32) & 0x01FFFFFFull) | (2u << 30);

        // D# group 1: data_size=2 (4B); tensor 8192x4, tile 8192x4, dim0_stride=8192
        const u32 tdim0 = 8192, tdim1 = 4, tile0 = 8192, tile1 = 4;
        const u64 stride0 = 8192;
        int32x8 g1;
        g1[0] = (int)(2u << 16);                                        // data_size
        g1[1] = (int)((tdim0 & 0xFFFFu) << 16);                         // tensor_dim0[15:0]
        g1[2] = (int)(((tdim0 >> 16) & 0xFFFFu) | ((tdim1 & 0xFFFFu) << 16));
        g1[3] = (int)(((tdim1 >> 16) & 0xFFFFu) | ((tile0 & 0xFFFFu) << 16));
        g1[4] = (int)(tile1 & 0xFFFFu);                                 // tile_dim1, tile_dim2=0
        g1[5] = (int)(stride0 & 0xFFFFFFFFull);                         // dim0_stride lo
        g1[6] = (int)((stride0 >> 32) & 0xFFFFull);                     // dim0_stride hi
        g1[7] = 0;

        int32x4 gz = {0, 0, 0, 0};
#if defined(__clang_major__) && (__clang_major__ >= 23)
        int32x8 gz8 = {0, 0, 0, 0, 0, 0, 0, 0};
        __builtin_amdgcn_tensor_load_to_lds(g0, g1, gz, gz, gz8, 0);
#else
        __builtin_amdgcn_tensor_load_to_lds(g0, g1, gz, gz, 0);
#endif
        __builtin_amdgcn_s_wait_tensorcnt(0);
    }
    __syncthreads();

    // --- candidate count N and total sum (keys are float bits, loss >= 0) ---
    u32 cnt = 0; float sall = 0.f;
    for (int i = threadIdx.x; i < AAA; i += 1024) {
        u32 kk = s_keys[i];
        if (kk) { cnt++; sall += __uint_as_float(kk); }
    }
    u32   N      = blk_red_u(cnt, s_ured);
    float sumAll = blk_red_f(sall, s_fred);

    int np = num_pos[lb];
    int kneed = NPPC * np;
    if (kneed < MIN_NEGC) kneed = MIN_NEGC;
    u32 ksel = (u32)kneed;

    float negsum;
    if (N <= ksel) {
        negsum = sumAll;                       // take all negatives
    } else {
        // 4-pass MSB-first radix select for the k-th largest key
        u32 prefix = 0, remaining = ksel;
        for (int pass = 0; pass < 4; ++pass) {
            int shift = 24 - 8 * pass;
            if (threadIdx.x < 256) s_hist[threadIdx.x] = 0;
            __syncthreads();
            u32 hiMask = (pass == 0) ? 0u : (0xFFFFFFFFu << (shift + 8));
            u32 hiVal  = (pass == 0) ? 0u : (prefix << (shift + 8));
            for (int i = threadIdx.x; i < AAA; i += 1024) {
                u32 kk = s_keys[i];
                if (kk && ((kk & hiMask) == hiVal))
                    atomicAdd(&s_hist[(kk >> shift) & 255u], 1u);
            }
            __syncthreads();
            if (threadIdx.x == 0) {
                u32 acc = 0; int chosen = 0;
                for (int b = 255; b >= 0; --b) {
                    u32 h = s_hist[b];
                    if (acc + h >= remaining) { chosen = b; break; }
                    acc += h;
                }
                s_bcast[0] = (prefix << 8) | (u32)chosen;
                s_bcast[1] = remaining - acc;      // in [1, hist[chosen]]
            }
            __syncthreads();
            prefix    = s_bcast[0];
            remaining = s_bcast[1];
            __syncthreads();
        }
        u32 T = prefix;                            // k-th largest key
        float sg = 0.f;
        for (int i = threadIdx.x; i < AAA; i += 1024) {
            u32 kk = s_keys[i];
            if (kk > T) sg += __uint_as_float(kk);
        }
        float SG = blk_red_f(sg, s_fred);
        // exact under ties: sum(>T) + T * (#slots left at the threshold)
        negsum = SG + __uint_as_float(T) * (float)remaining;
    }

    if (threadIdx.x == 0) {
        totals[lb] = pos_sum[lb] + negsum;         // one block per lb: plain store
    }
}

// ---------------------------------------------------------------- finalize
// Single wave. Batch reduction done with V_WMMA_F32_16X16X4_F32 (exact f32):
// D = A x ones(4x16) gives row sums; each level's 32 totals occupy 8 rows of
// one A operand, so summing the 8 D VGPRs in a lane yields the level sum.
//   WMMA#0: rows 0-7  -> level 0 (lanes 0-15), rows 8-15 -> level 1 (lanes 16-31)
//   WMMA#1: rows 0-7  -> level 2 (lanes 0-15), rows 8-15 -> zero padding
__global__ __launch_bounds__(32) void k_final(const float* __restrict__ totals,
                                              const int* __restrict__ num_pos,
                                              float* __restrict__ out) {
    int l = threadIdx.x & 31;
    int mrow  = l & 15;
    int kbase = (l < 16) ? 0 : 2;     // A 16x4 f32 layout: lanes 0-15 K0/K1, 16-31 K2/K3

    v2f a0, a1, b;
    a0.x = totals[mrow * 4 + kbase + 0];
    a0.y = totals[mrow * 4 + kbase + 1];
    a1.x = totals[64 + mrow * 4 + kbase + 0];
    a1.y = totals[64 + mrow * 4 + kbase + 1];
    b.x = 1.0f; b.y = 1.0f;           // ones matrix: layout-agnostic

    v8f c = {};
    v8f d0 = __builtin_amdgcn_wmma_f32_16x16x4_f32(
        false, a0, false, b, (short)0, c, false, false);
    v8f d1 = __builtin_amdgcn_wmma_f32_16x16x4_f32(
        false, a1, false, b, (short)0, c, false, false);

    float s0 = d0[0] + d0[1] + d0[2] + d0[3] + d0[4] + d0[5] + d0[6] + d0[7];
    float s1 = d1[0] + d1[1] + d1[2] + d1[3] + d1[4] + d1[5] + d1[6] + d1[7];

    float lev0 = __shfl(s0, 0, 32);   // rows 0-7   = totals[ 0..31]
    float lev1 = __shfl(s0, 16, 32);  // rows 8-15  = totals[32..63]
    float lev2 = __shfl(s1, 0, 32);   // rows 0-7   = totals[64..95]

    if (l == 0) {
        float lev[3] = {lev0, lev1, lev2};
        float acc = 0.f;
        for (int lv = 0; lv < LVL; ++lv) {
            int np = 0;
            for (int bb = 0; bb < BB; ++bb) np += num_pos[lv * BB + bb];
            acc += lev[lv] / fmaxf(1.0f, (float)np);
        }
        out[0] = acc / (float)LVL;
    }
}

// ---------------------------------------------------------------- launch
extern "C" void kernel_launch(void* const* d_in, const int* in_sizes, int n_in,
                              void* d_out, int out_size, void* d_ws, size_t ws_size,
                              hipStream_t stream) {
    (void)in_sizes; (void)n_in; (void)out_size; (void)ws_size;
    const float* logits = (const float*)d_in[0];
    const int*   labels = (const int*)d_in[1];
    const float* tags   = (const float*)d_in[2];
    float* out = (float*)d_out;

    char*  ws        = (char*)d_ws;
    u32*   negkey    = (u32*)ws;
    float* pos_sum   = (float*)(ws + NEGKEY_BYTES);
    int*   num_pos   = (int*)(ws + NEGKEY_BYTES + 512);
    float* totals    = (float*)(ws + NEGKEY_BYTES + 1024);   // 128 floats, padded

    k_init<<<1, 256, 0, stream>>>(pos_sum, num_pos, totals);
    int blocks = (int)(TOTAL_ANCH / 8);   // 8 waves (=8 anchors) per 256-thread block
    k_loss<<<blocks, 256, 0, stream>>>(logits, labels, tags, negkey, pos_sum, num_pos);
    k_mine<<<NLB, 1024, 0, stream>>>(negkey, pos_sum, num_pos, totals);
    k_final<<<1, 32, 0, stream>>>(totals, num_pos, out);
}